// RankLoss_73959336837390
// MI455X (gfx1250) — compile-verified
//
#include <hip/hip_runtime.h>

typedef float v2f __attribute__((ext_vector_type(2)));
typedef float v4f __attribute__((ext_vector_type(4)));
typedef float v8f __attribute__((ext_vector_type(8)));

#define RL_B 32
#define RL_N 2048
#define RL_LOG2E 1.44269504088896340736f

// sigmoid(xv - x0) = 1 / (1 + exp2(fma(xv, -log2e, x0*log2e)))
// -> v_fma_f32 (literal + SGPR), v_exp_f32, v_add_f32, v_rcp_f32 per element.
__device__ __forceinline__ float fast_sigmoid_exp2(float xv, float k) {
    return __builtin_amdgcn_rcpf(
        1.0f + __builtin_amdgcn_exp2f(__builtin_fmaf(xv, -RL_LOG2E, k)));
}

// One wave (32 lanes, EXEC all-1s) per batch row b.
// s_b = sum_j sigmoid(x[b,j] - x[b,0]); the reduction runs on the matrix pipe:
//   V_WMMA_F32_16X16X4_F32 with B = ones  ->  D = rowsum(A) + C  (exact f32).
// Each b128 load feeds two WMMA steps (128 elements / wave / iteration);
// exp/rcp ride the transcendental pipe and co-issue against the WMMAs.
__global__ __launch_bounds__(32)
void rankloss_rowsum_wmma(const float* __restrict__ x, float* __restrict__ rr_ws) {
    const int b    = blockIdx.x;          // 0..31
    const int lane = threadIdx.x;         // 0..31 (wave32)
    const float* row = x + (size_t)b * RL_N;
    const float x0 = row[0];              // uniform -> scalar load
    const float k  = x0 * RL_LOG2E;       // uniform -> SGPR

    v8f acc = {};                         // C/D accumulator (16x16 f32)
    const v2f ones = {1.0f, 1.0f};        // B matrix = ones(4x16)

    for (int base = 0; base < RL_N; base += 128) {
        const v4f xv = *(const v4f*)(row + base + 4 * lane);  // global_load_b128
        v2f a0, a1;
        a0.x = fast_sigmoid_exp2(xv.x, k);
        a0.y = fast_sigmoid_exp2(xv.y, k);
        a1.x = fast_sigmoid_exp2(xv.z, k);
        a1.y = fast_sigmoid_exp2(xv.w, k);
        acc = __builtin_amdgcn_wmma_f32_16x16x4_f32(
            false, a0, false, ones, (short)0, acc, false, false);
        acc = __builtin_amdgcn_wmma_f32_16x16x4_f32(
            false, a1, false, ones, (short)0, acc, false, false);
    }

    // D[i][j] = rowsum_i replicated across all 16 columns:
    //   lanes 0-15  (VGPRs 0-7) -> rows M=0..7, lanes 16-31 -> rows M=8..15.
    float part = acc[0] + acc[1] + acc[2] + acc[3] +
                 acc[4] + acc[5] + acc[6] + acc[7];
    part += __shfl_xor(part, 16, 32);     // lane 0: s_b over all 16 rows

    if (lane == 0)
        rr_ws[b] = 1.0f / (part + 0.5f);  // rr[b,0] (precise; once per wave)
}

// Deterministic final reduction of the 32 per-batch reciprocal ranks.
__global__ __launch_bounds__(32)
void rankloss_finalize(const float* __restrict__ rr_ws, float* __restrict__ out) {
    float v = rr_ws[threadIdx.x];         // exactly 32 values
    #pragma unroll
    for (int off = 16; off > 0; off >>= 1)
        v += __shfl_xor(v, off, 32);
    if (threadIdx.x == 0)
        out[0] = -v * (1.0f / (float)RL_B);
}

extern "C" void kernel_launch(void* const* d_in, const int* in_sizes, int n_in,
                              void* d_out, int out_size, void* d_ws, size_t ws_size,
                              hipStream_t stream) {
    (void)in_sizes; (void)n_in; (void)out_size; (void)ws_size;
    const float* x   = (const float*)d_in[0];  // [32, 2048] f32
    float*       out = (float*)d_out;          // scalar loss
    float*       ws  = (float*)d_ws;           // 32 floats of scratch

    rankloss_rowsum_wmma<<<RL_B, 32, 0, stream>>>(x, ws);
    rankloss_finalize<<<1, 32, 0, stream>>>(ws, out);
}